// MultiHeadAttnRelative_55980603736096
// MI455X (gfx1250) — compile-verified
//
#include <hip/hip_runtime.h>
#include <hip/hip_bf16.h>
#include <math.h>

// ---------------------------------------------------------------------------
// MI455X (gfx1250) fused relative-position multi-head attention.
//
// Roofline: relative_k + relative_v = 1 GB of mandatory one-touch HBM reads
// -> ~43 us floor at 23.3 TB/s. All GEMMs together are ~1.6 GFLOP and all
// non-relative tensors (<8 MB) are L2-resident, so the design goal is
// perfectly-coalesced single-pass streaming of the relative tensors with the
// logits/softmax/context math fused on top, plus WMMA for the dense GEMMs.
// ---------------------------------------------------------------------------

typedef __attribute__((ext_vector_type(2))) float v2f;
typedef __attribute__((ext_vector_type(8))) float v8f;

#define L_SEQ 512
#define DMODEL 512
#define HEADS 8
#define DHEAD 64

// ---------------------------------------------------------------------------
// Dense GEMM: C[512x512] = (A[512x512] @ W[512x512] + bias) * scale
// One 16x16 output tile per wave via V_WMMA_F32_16X16X4_F32 (exact f32).
// 1024 tiles total -> 128 blocks x 8 waves.
// ---------------------------------------------------------------------------
__global__ __launch_bounds__(256) void gemm512_bias_wmma(
    const float* __restrict__ A, const float* __restrict__ W,
    const float* __restrict__ bias, float* __restrict__ C, float scale)
{
    const int lane   = threadIdx.x & 31;
    const int waveId = (blockIdx.x * (blockDim.x >> 5)) + (threadIdx.x >> 5);
    const int tm   = waveId >> 5;     // 0..31 tile row
    const int tn   = waveId & 31;     // 0..31 tile col
    const int half = lane >> 4;       // lane[4] selects K-pair (ISA 7.12.2)
    const int l16  = lane & 15;

    const int mA = tm * 16 + l16;     // A-fragment row for this lane
    const int nB = tn * 16 + l16;     // B/C-fragment column for this lane

    v8f acc0 = {};
    v8f acc1 = {};

    const float* Arow = A + (size_t)mA * DMODEL;

    for (int k = 0; k < DMODEL; k += 8) {
        // ---- K-step 0 (k..k+3) ----
        {
            const int kb = k + 2 * half;
            v2f a = *(const v2f*)(Arow + kb);              // A[m, kb], A[m, kb+1]
            v2f b;
            b.x = W[(size_t)kb * DMODEL + nB];             // B[kb,   n]
            b.y = W[(size_t)(kb + 1) * DMODEL + nB];       // B[kb+1, n]
            acc0 = __builtin_amdgcn_wmma_f32_16x16x4_f32(
                false, a, false, b, (short)0, acc0, false, false);
        }
        // ---- K-step 1 (k+4..k+7), independent accumulator for ILP ----
        {
            const int kb = k + 4 + 2 * half;
            v2f a = *(const v2f*)(Arow + kb);
            v2f b;
            b.x = W[(size_t)kb * DMODEL + nB];
            b.y = W[(size_t)(kb + 1) * DMODEL + nB];
            acc1 = __builtin_amdgcn_wmma_f32_16x16x4_f32(
                false, a, false, b, (short)0, acc1, false, false);
        }
    }

    const v8f acc = acc0 + acc1;
    const float bn = bias[nB];
#pragma unroll
    for (int r = 0; r < 8; ++r) {
        const int m = tm * 16 + r + 8 * half;              // C layout, ISA 7.12.2
        C[(size_t)m * DMODEL + nB] = (acc[r] + bn) * scale;
    }
}

// ---------------------------------------------------------------------------
// Fused relative attention. One block per (l, h); 256 threads = 8 waves.
//  Phase A: logits[m] = sum_d q[l,d] * (k[m,d] + rel_k[l,m,d])   (shfl reduce)
//  Phase B: softmax over m (deterministic LDS tree), write attn output
//  Phase C: ctx[d]   = sum_m attn[m] * (v[m,d] + rel_v[l,m,d])
// Each lane owns d-pair (2*lane, 2*lane+1) -> every rel_k/rel_v row read is a
// single fully-coalesced 256B wave burst; the 128KB-per-(h,l) contiguous
// rel slices stream at full HBM rate.
// ---------------------------------------------------------------------------
__global__ __launch_bounds__(256) void attn_rel_fused(
    const float* __restrict__ Q, const float* __restrict__ K,
    const float* __restrict__ V,
    const float* __restrict__ relk, const float* __restrict__ relv,
    float* __restrict__ attn_out, float* __restrict__ ctx_out)
{
    const int l    = blockIdx.x;          // 0..511 query position
    const int h    = blockIdx.y;          // 0..7 head
    const int tid  = threadIdx.x;         // 0..255
    const int wave = tid >> 5;            // 0..7
    const int lane = tid & 31;

    __shared__ float sLogit[L_SEQ];
    __shared__ float sRed[256];
    __shared__ float sPart[8][DHEAD];

    // q d-pair for this lane (already scaled by 1/sqrt(D) in projection)
    const float2 qd = *(const float2*)(Q + (size_t)l * DMODEL + h * DHEAD + 2 * lane);

    const float* kcol = K + h * DHEAD + 2 * lane;                      // + m*512
    const size_t relbase = ((size_t)(h * L_SEQ + l)) * L_SEQ * DHEAD;  // [h][l][m][d]
    const float* rk = relk + relbase + 2 * lane;

    // ---------------- Phase A: logits ----------------
    const int m0 = wave * 64;
#pragma unroll 4
    for (int mi = 0; mi < 64; ++mi) {
        const int m = m0 + mi;
        const float2 kk = *(const float2*)(kcol + (size_t)m * DMODEL);  // L2 hit
        const float2 rr = *(const float2*)(rk + (size_t)m * DHEAD);     // HBM stream
        float p = qd.x * (kk.x + rr.x) + qd.y * (kk.y + rr.y);
        p += __shfl_xor(p, 16, 32);
        p += __shfl_xor(p, 8, 32);
        p += __shfl_xor(p, 4, 32);
        p += __shfl_xor(p, 2, 32);
        p += __shfl_xor(p, 1, 32);
        if (lane == 0) sLogit[m] = p;
    }
    __syncthreads();

    // ---------------- Phase B: softmax ----------------
    const float a0 = sLogit[tid];
    const float a1 = sLogit[tid + 256];
    sRed[tid] = fmaxf(a0, a1);
    __syncthreads();
    for (int s = 128; s > 0; s >>= 1) {
        if (tid < s) sRed[tid] = fmaxf(sRed[tid], sRed[tid + s]);
        __syncthreads();
    }
    const float mx = sRed[0];
    __syncthreads();

    float e0 = __expf(a0 - mx);
    float e1 = __expf(a1 - mx);
    sRed[tid] = e0 + e1;
    __syncthreads();
    for (int s = 128; s > 0; s >>= 1) {
        if (tid < s) sRed[tid] += sRed[tid + s];
        __syncthreads();
    }
    const float inv = 1.0f / sRed[0];
    __syncthreads();

    e0 *= inv;
    e1 *= inv;
    sLogit[tid]       = e0;   // reuse as attn weights for phase C
    sLogit[tid + 256] = e1;
    __syncthreads();

    // coalesced attention-weights output (second tuple element)
    float* ao = attn_out + ((size_t)(h * L_SEQ + l)) * L_SEQ;
    ao[tid]       = e0;
    ao[tid + 256] = e1;

    // ---------------- Phase C: context ----------------
    const float* vcol = V + h * DHEAD + 2 * lane;
    const float* rv   = relv + relbase + 2 * lane;
    float2 acc = make_float2(0.0f, 0.0f);
#pragma unroll 4
    for (int mi = 0; mi < 64; ++mi) {
        const int m = m0 + mi;
        const float a = sLogit[m];
        const float2 vv = *(const float2*)(vcol + (size_t)m * DMODEL);  // L2 hit
        const float2 rr = *(const float2*)(rv + (size_t)m * DHEAD);     // HBM stream
        acc.x = fmaf(a, vv.x + rr.x, acc.x);
        acc.y = fmaf(a, vv.y + rr.y, acc.y);
    }
    sPart[wave][2 * lane]     = acc.x;
    sPart[wave][2 * lane + 1] = acc.y;
    __syncthreads();

    if (tid < DHEAD) {
        // fixed-order cross-wave reduction: bitwise deterministic
        float s = sPart[0][tid];
#pragma unroll
        for (int w = 1; w < 8; ++w) s += sPart[w][tid];
        ctx_out[(size_t)l * DMODEL + h * DHEAD + tid] = s;
    }
}

// ---------------------------------------------------------------------------
// Launch
// inputs: 0 query, 1 key, 2 value, 3 relative_k, 4 relative_v,
//         5 Wq, 6 bq, 7 Wk, 8 bk, 9 Wv, 10 bv, 11 Wo, 12 bo
// output: [output (512*512) | attn (8*512*512)]
// ---------------------------------------------------------------------------
extern "C" void kernel_launch(void* const* d_in, const int* in_sizes, int n_in,
                              void* d_out, int out_size, void* d_ws, size_t ws_size,
                              hipStream_t stream)
{
    const float* query = (const float*)d_in[0];
    const float* key   = (const float*)d_in[1];
    const float* value = (const float*)d_in[2];
    const float* relk  = (const float*)d_in[3];
    const float* relv  = (const float*)d_in[4];
    const float* Wq = (const float*)d_in[5];
    const float* bq = (const float*)d_in[6];
    const float* Wk = (const float*)d_in[7];
    const float* bk = (const float*)d_in[8];
    const float* Wv = (const float*)d_in[9];
    const float* bv = (const float*)d_in[10];
    const float* Wo = (const float*)d_in[11];
    const float* bo = (const float*)d_in[12];

    float* out      = (float*)d_out;                       // [512,512] projected output
    float* attn_out = out + (size_t)L_SEQ * DMODEL;        // [8,512,512] attention weights

    float* w   = (float*)d_ws;
    float* Qb  = w;                                        // 1 MB each
    float* Kb  = w + 1 * (size_t)L_SEQ * DMODEL;
    float* Vb  = w + 2 * (size_t)L_SEQ * DMODEL;
    float* Ctx = w + 3 * (size_t)L_SEQ * DMODEL;

    const float qscale = 0.125f;                           // 1/sqrt(DHEAD)

    dim3 gblk(128), gthr(256);
    hipLaunchKernelGGL(gemm512_bias_wmma, gblk, gthr, 0, stream, query, Wq, bq, Qb, qscale);
    hipLaunchKernelGGL(gemm512_bias_wmma, gblk, gthr, 0, stream, key,   Wk, bk, Kb, 1.0f);
    hipLaunchKernelGGL(gemm512_bias_wmma, gblk, gthr, 0, stream, value, Wv, bv, Vb, 1.0f);

    dim3 ablk(L_SEQ, HEADS), athr(256);
    hipLaunchKernelGGL(attn_rel_fused, ablk, athr, 0, stream,
                       Qb, Kb, Vb, relk, relv, attn_out, Ctx);

    hipLaunchKernelGGL(gemm512_bias_wmma, gblk, gthr, 0, stream, Ctx, Wo, bo, out, 1.0f);
}